// FenwickTreeDeformation_7687991460416
// MI455X (gfx1250) — compile-verified
//
#include <hip/hip_runtime.h>

typedef float v2f __attribute__((ext_vector_type(2)));
typedef float v8f __attribute__((ext_vector_type(8)));

#define T_FRAMES 1024
#define NCOL_T (4096 * 3)   // translations as [1025 x 12288] f32
#define NCOL_R (4096 * 4)   // rotations    as [1025 x 16384] f32

// c is on the within-block Fenwick chain of r iff exists j: c == (r>>j)<<j.
constexpr bool chain_has(int r, int c) {
    for (int j = 0; j <= 4; ++j)
        if (((r >> j) << j) == c) return true;
    return false;
}

// Per-lane A-operand constants for V_WMMA_F32_16X16X4_F32:
// lane L: row M = L&15; VGPR0 holds K = ch*4 + (L>=16?2:0), VGPR1 holds K+1.
struct LTab { float v[32][8]; };
constexpr LTab make_ltab() {
    LTab t{};
    for (int lane = 0; lane < 32; ++lane) {
        const int n = lane & 15, hi = lane >> 4;
        for (int ch = 0; ch < 4; ++ch) {
            const int k0 = ch * 4 + hi * 2;
            t.v[lane][2 * ch]     = chain_has(n, k0) ? 1.f : 0.f;
            t.v[lane][2 * ch + 1] = chain_has(n, k0 + 1) ? 1.f : 0.f;
        }
    }
    return t;
}
__constant__ LTab LTAB = make_ltab();

// Level-1: S[m] = node[16m] + S[m - lowbit(m)], m = 1..64, written raw into
// output rows 16m-1 (they are the t=16m outputs; the WMMA kernel reads them
// back as X row 0 and overwrites them, normalized for quaternions).
template <int NCOL>
__global__ void fenwick_bases(const float* __restrict__ node,
                              float* __restrict__ out) {
    const int col = blockIdx.x * blockDim.x + threadIdx.x;  // NCOL % 256 == 0
    float st[7] = {0.f, 0.f, 0.f, 0.f, 0.f, 0.f, 0.f};
#pragma unroll
    for (int m = 1; m <= 64; ++m) {
        const int b = __builtin_ctz(m);
        const float acc = st[b] + node[(size_t)(16 * m) * NCOL + col];
#pragma unroll
        for (int j = 0; j <= b; ++j) st[j] = acc;
        out[(size_t)(16 * m - 1) * NCOL + col] = acc;
    }
}

// Level-0: per block m and 32-column pair tile, D = L x X via chained
// v_wmma_f32_16x16x4_f32 (A shared by both 16-col tiles).  X row k (k>=1) =
// node[16m+k], X row 0 = S[m] (read back from out).  m=1..63 is a guard-free
// fast path with all row offsets folded to immediates; m=0 / m=64 take the
// guarded edge path.
template <int NCOL, bool NORM>
__global__ void __launch_bounds__(256)
fenwick_wmma(const float* __restrict__ node, float* __restrict__ out) {
    const int lane = threadIdx.x & 31;
    const int wavei = threadIdx.x >> 5;
    const int pair = blockIdx.x * 8 + wavei;      // 32-column pair tile
    const int m = blockIdx.y;                     // 0..64
    const int n = lane & 15;
    const int hi = lane >> 4;
    const int c0 = pair * 32 + n;                 // tile0 column; tile1 = c0+16

    // A operand from constant table (2 x b128 per lane)
    v2f a[4];
    {
        const float* lp = LTAB.v[lane];
#pragma unroll
        for (int ch = 0; ch < 4; ++ch) {
            a[ch].x = lp[2 * ch];
            a[ch].y = lp[2 * ch + 1];
        }
    }

    const bool fast = (m >= 1) && (m <= 63);
    const size_t rowb = (size_t)(16 * m);

    v2f b0[4], b1[4];
    if (fast) {
        const float* nbase = node + (rowb + (size_t)(hi * 2)) * NCOL + c0;
        const float* sbase = out + (rowb - 1) * NCOL + c0;
        const float* p0 = (hi == 0) ? sbase : nbase;  // ch0 .x source select
#pragma unroll
        for (int ch = 0; ch < 4; ++ch) {
            const int off = ch * 4 * NCOL;            // immediate-foldable
            b0[ch].x = (ch == 0) ? p0[0]
                                 : __builtin_nontemporal_load(nbase + off);
            b0[ch].y = __builtin_nontemporal_load(nbase + off + NCOL);
            b1[ch].x = (ch == 0) ? p0[16]
                                 : __builtin_nontemporal_load(nbase + off + 16);
            b1[ch].y = __builtin_nontemporal_load(nbase + off + NCOL + 16);
        }
    } else {  // m == 0 or m == 64
#pragma unroll
        for (int ch = 0; ch < 4; ++ch) {
            const int k0 = ch * 4 + hi * 2;
            float x0a, x0b, x1a, x1b;
            if (k0 == 0) {
                if (m == 0) { x0a = 0.f; x0b = 0.f; }
                else {
                    x0a = out[(rowb - 1) * NCOL + c0];
                    x0b = out[(rowb - 1) * NCOL + c0 + 16];
                }
            } else {
                const int row = 16 * m + k0;
                const bool ok = row <= T_FRAMES;
                x0a = ok ? node[(size_t)row * NCOL + c0] : 0.f;
                x0b = ok ? node[(size_t)row * NCOL + c0 + 16] : 0.f;
            }
            {
                const int row = 16 * m + k0 + 1;  // k >= 1 always
                const bool ok = row <= T_FRAMES;
                x1a = ok ? node[(size_t)row * NCOL + c0] : 0.f;
                x1b = ok ? node[(size_t)row * NCOL + c0 + 16] : 0.f;
            }
            b0[ch].x = x0a; b0[ch].y = x1a;
            b1[ch].x = x0b; b1[ch].y = x1b;
        }
    }

    v8f acc0 = {}, acc1 = {};
#pragma unroll
    for (int ch = 0; ch < 4; ++ch) {
        acc0 = __builtin_amdgcn_wmma_f32_16x16x4_f32(
            false, a[ch], false, b0[ch], (short)0, acc0, false, false);
        acc1 = __builtin_amdgcn_wmma_f32_16x16x4_f32(
            false, a[ch], false, b1[ch], (short)0, acc1, false, false);
    }

    // Optional quaternion normalization (components = 4 adjacent lanes)
    float r0[8], r1[8];
#pragma unroll
    for (int v = 0; v < 8; ++v) {
        float x0 = acc0[v], x1 = acc1[v];
        if (NORM) {
            float na = x0 * x0, nb = x1 * x1;
            na += __shfl_xor(na, 1, 32);
            nb += __shfl_xor(nb, 1, 32);
            na += __shfl_xor(na, 2, 32);
            nb += __shfl_xor(nb, 2, 32);
            x0 *= rsqrtf(fmaxf(na, 1e-24f));   // == x0 / max(||q||, 1e-12)
            x1 *= rsqrtf(fmaxf(nb, 1e-24f));
        }
        r0[v] = x0; r1[v] = x1;
    }

    if (fast) {
        float* sb = out + (rowb - 1 + (size_t)(hi * 8)) * NCOL + c0;
#pragma unroll
        for (int v = 0; v < 8; ++v) {          // immediate-offset stores
            __builtin_nontemporal_store(r0[v], sb + v * NCOL);
            __builtin_nontemporal_store(r1[v], sb + v * NCOL + 16);
        }
    } else {
#pragma unroll
        for (int v = 0; v < 8; ++v) {
            const int t = 16 * m + v + hi * 8;
            if (t >= 1 && t <= T_FRAMES) {
                out[(size_t)(t - 1) * NCOL + c0] = r0[v];
                out[(size_t)(t - 1) * NCOL + c0 + 16] = r1[v];
            }
        }
    }
}

extern "C" void kernel_launch(void* const* d_in, const int* in_sizes, int n_in,
                              void* d_out, int out_size, void* d_ws, size_t ws_size,
                              hipStream_t stream) {
    const float* nt = (const float*)d_in[0];  // [1025, 4096, 3] f32
    const float* nr = (const float*)d_in[1];  // [1025, 4096, 4] f32
    float* out_t = (float*)d_out;                          // [1024, 4096, 3]
    float* out_r = out_t + (size_t)T_FRAMES * NCOL_T;      // [1024, 4096, 4]

    fenwick_bases<NCOL_T><<<NCOL_T / 256, 256, 0, stream>>>(nt, out_t);
    fenwick_bases<NCOL_R><<<NCOL_R / 256, 256, 0, stream>>>(nr, out_r);

    dim3 gt(NCOL_T / 32 / 8, 65);   // 48 x 65
    fenwick_wmma<NCOL_T, false><<<gt, 256, 0, stream>>>(nt, out_t);
    dim3 gr(NCOL_R / 32 / 8, 65);   // 64 x 65
    fenwick_wmma<NCOL_R, true><<<gr, 256, 0, stream>>>(nr, out_r);
}